// MargExcLoss_78537771974698
// MI455X (gfx1250) — compile-verified
//
#include <hip/hip_runtime.h>
#include <hip/hip_bf16.h>
#include <math.h>

typedef __attribute__((ext_vector_type(2))) float v2f;
typedef __attribute__((ext_vector_type(8))) float v8f;

#define NVOX   524288      // D*H*W = 32*128*128
#define NBATCH 4
#define NBLK   256         // blocks per batch item
#define NEGV   (-1.0e9f)
#define SMOOTHV 1e-5f

// 32 -> 16 lane fold in one matrix op.
// V_WMMA_F32_16X16X4_F32 with A = ones(16x4), B = {v, 0}:
//   D[m,n] = sum_k B[k,n] = v[n] + v[n+16]   (n = lane & 15)
// Layout-robust: the two lane halves of b0 are two distinct K rows whatever
// their labels; b1 = 0 kills the remaining K rows; A = ones is layout-free.
// Result: every lane holds v[l&15] + v[(l&15)+16]; lanes 0..15 carry the
// 16 partial sums.
__device__ __forceinline__ float wmma_fold32_16(float v) {
  v2f ones; ones[0] = 1.0f; ones[1] = 1.0f;
  v2f bm;   bm[0] = v;      bm[1] = 0.0f;
  v8f z = {};
  v8f d = __builtin_amdgcn_wmma_f32_16x16x4_f32(
      /*neg_a=*/false, ones, /*neg_b=*/false, bm,
      /*c_mod=*/(short)0, z, /*reuse_a=*/false, /*reuse_b=*/false);
  return d[0];
}

// Partial slot layout per (b, blk), 64 floats:
//  [0..11]  sum p_c * te        [12..23] sum p_c^2
//  [24..35] sum log1p(p_c)*te   [36..47] sum_{tgt==c} p_c
//  [48..59] count(tgt==c)       [60] nll sum   [61..63] zero pad
__global__ __launch_bounds__(256) void marg_exc_main(
    const float* __restrict__ inputs, const int* __restrict__ targets,
    const int* __restrict__ task_id, float* __restrict__ partials) {
  const int b   = blockIdx.y;
  const int blk = blockIdx.x;
  const int tid = threadIdx.x;

  const int t    = task_id[b];
  const int ids1 = (t <= 3) ? (2 * t + 1) : (t + 5);   // NBG[t][1]
  const int ids2 = (t <= 3) ? (2 * t + 2) : -1;        // NBG[t][2]
  const int dlt  = (t <= 4) ? (-2 * t) : (-t - 4);

  const float* __restrict__ inb = inputs + (size_t)b * 12u * NVOX;
  const int*   __restrict__ tg  = targets + (size_t)b * NVOX;

  float a_pte[12], a_p2[12], a_lte[12], a_pc[12], a_cnt[12];
  #pragma unroll
  for (int c = 0; c < 12; ++c) {
    a_pte[c] = 0.f; a_p2[c] = 0.f; a_lte[c] = 0.f; a_pc[c] = 0.f; a_cnt[c] = 0.f;
  }
  float a_nll = 0.f;

  for (int idx = blk * 256 + tid; idx < NVOX; idx += NBLK * 256) {
    const int tv = __builtin_nontemporal_load(&tg[idx]);
    float x[12];
    #pragma unroll
    for (int c = 0; c < 12; ++c)
      x[c] = __builtin_nontemporal_load(&inb[(size_t)c * NVOX + idx]);

    float m = x[0];
    #pragma unroll
    for (int c = 1; c < 12; ++c) m = fmaxf(m, x[c]);
    float e[12]; float s = 0.f;
    #pragma unroll
    for (int c = 0; c < 12; ++c) { e[c] = __expf(x[c] - m); s += e[c]; }
    const float inv = 1.0f / s;

    float g0 = NEGV, g1 = NEGV, g2 = NEGV;
    #pragma unroll
    for (int c = 0; c < 12; ++c) {
      const float p = e[c] * inv;
      a_p2[c] += p * p;
      const bool eq = (tv == c);
      a_pc[c]  += eq ? p   : 0.f;
      a_cnt[c] += eq ? 1.f : 0.f;
      const float l = log1pf(p);
      const bool te = (c != 0) && !eq;
      a_pte[c] += te ? p : 0.f;
      a_lte[c] += te ? l : 0.f;
      if (c == 0)    g0 = p;
      if (c == ids1) g1 = p;
      if (c == ids2) g2 = p;    // never taken when ids2 == -1 -> g2 stays NEG
    }
    // 3-way log-softmax NLL over (g0, g1, g2)
    const float m3  = fmaxf(g0, fmaxf(g1, g2));
    const float lse = __logf(__expf(g0 - m3) + __expf(g1 - m3) + __expf(g2 - m3));
    const int   tm  = (tv > 0) ? (tv + dlt) : 0;     // local slot 0..2
    const float gs  = (tm == 0) ? g0 : ((tm == 1) ? g1 : g2);
    a_nll += (m3 + lse) - gs;
  }

  // ---- deterministic block reduction ----
  // WMMA folds 32 lanes -> 16 partials; lanes 0..15 store them straight to
  // LDS (no ds_bpermute round trips); 64 threads then sum 8 waves x 16 lanes
  // per slot in fixed order with float4 LDS loads.
  __shared__ float lred[8][64][16];                // 32 KB
  const int wave = tid >> 5;
  const int lane = tid & 31;
  float s;
  #pragma unroll
  for (int c = 0; c < 12; ++c) {
    s = wmma_fold32_16(a_pte[c]); if (lane < 16) lred[wave][c][lane]      = s;
    s = wmma_fold32_16(a_p2[c]);  if (lane < 16) lred[wave][12 + c][lane] = s;
    s = wmma_fold32_16(a_lte[c]); if (lane < 16) lred[wave][24 + c][lane] = s;
    s = wmma_fold32_16(a_pc[c]);  if (lane < 16) lred[wave][36 + c][lane] = s;
    s = wmma_fold32_16(a_cnt[c]); if (lane < 16) lred[wave][48 + c][lane] = s;
  }
  s = wmma_fold32_16(a_nll);
  if (lane < 16) {
    lred[wave][60][lane] = s;
    lred[wave][61][lane] = 0.f; lred[wave][62][lane] = 0.f; lred[wave][63][lane] = 0.f;
  }
  __syncthreads();
  if (tid < 64) {
    float acc = 0.f;
    #pragma unroll
    for (int w = 0; w < 8; ++w) {
      const float4* q = (const float4*)(&lred[w][tid][0]);
      const float4 q0 = q[0], q1 = q[1], q2 = q[2], q3 = q[3];
      acc += ((q0.x + q0.y) + (q0.z + q0.w)) + ((q1.x + q1.y) + (q1.z + q1.w))
           + ((q2.x + q2.y) + (q2.z + q2.w)) + ((q3.x + q3.y) + (q3.z + q3.w));
    }
    partials[(((size_t)b * NBLK + blk) << 6) + tid] = acc;
  }
}

// Single-wave finalize: wmma chains (A = ones, C accumulate) sum 4 block
// partials per instruction across 16 slots in parallel, deterministically.
__global__ __launch_bounds__(32) void marg_exc_finalize(
    const float* __restrict__ partials, const int* __restrict__ task_id,
    float* __restrict__ out) {
  const int lane = threadIdx.x;            // 0..31, one full wave32
  __shared__ float red[NBATCH][64];
  v2f ones; ones[0] = 1.0f; ones[1] = 1.0f;
  const int half = lane >> 4;
  const int col  = lane & 15;
  for (int b = 0; b < NBATCH; ++b) {
    for (int g = 0; g < 4; ++g) {
      v8f acc = {};
      for (int it = 0; it < NBLK / 4; ++it) {
        const int blkA = it * 4 + half * 2;   // lanes<16: blk,blk+1; lanes>=16: blk+2,blk+3
        const size_t base = (((size_t)b * NBLK + blkA) << 6) + g * 16 + col;
        v2f bm; bm[0] = partials[base]; bm[1] = partials[base + 64];
        acc = __builtin_amdgcn_wmma_f32_16x16x4_f32(
            false, ones, false, bm, (short)0, acc, false, false);
      }
      if (lane < 16) red[b][g * 16 + lane] = acc[0];   // D[0,n], n = lane
    }
  }
  __syncthreads();
  if (lane == 0) {
    const float Nf = (float)NVOX;
    float L_md = 0.f, L_mc = 0.f, L_ed = 0.f, L_ec = 0.f;
    for (int b = 0; b < NBATCH; ++b) {
      const float* r = red[b];
      const int t    = task_id[b];
      const int ids1 = (t <= 3) ? (2 * t + 1) : (t + 5);
      const int ids2 = (t <= 3) ? (2 * t + 2) : -1;
      L_mc += r[60] / Nf;                               // mean NLL per b
      int jid[3] = {0, ids1, ids2};
      const int nj = (ids2 >= 0) ? 3 : 2;               // masked slot -> dice=1 -> 0
      for (int j = 0; j < nj; ++j) {
        const int id = jid[j];
        const float inter = r[36 + id], y = r[48 + id], z = r[12 + id];
        L_md += 1.f - (2.f * inter + SMOOTHV) / (z + y + SMOOTHV);
      }
      for (int c = 0; c < 12; ++c) {
        const float ye = (c == 0) ? 0.f : (Nf - r[48 + c]);
        L_ed += (2.f * r[c] + SMOOTHV) / (r[12 + c] + ye + SMOOTHV);
        L_ec += r[24 + c] / Nf;
      }
    }
    out[0] = L_md * 0.25f;   // loss_marg_dice
    out[1] = L_mc * 0.25f;   // loss_marg_ce
    out[2] = L_ed * 0.25f;   // loss_exc_dice
    out[3] = L_ec * 0.25f;   // loss_exc_ce
  }
}

extern "C" void kernel_launch(void* const* d_in, const int* in_sizes, int n_in,
                              void* d_out, int out_size, void* d_ws, size_t ws_size,
                              hipStream_t stream) {
  (void)in_sizes; (void)n_in; (void)out_size; (void)ws_size;
  const float* inputs  = (const float*)d_in[0];   // (4,12,32,128,128) f32
  const int*   targets = (const int*)d_in[1];     // (4,32,128,128) i32
  const int*   task_id = (const int*)d_in[2];     // (4,) i32
  float* out      = (float*)d_out;                // 4 scalars
  float* partials = (float*)d_ws;                 // NBATCH*NBLK*64 f32 = 256 KB

  marg_exc_main<<<dim3(NBLK, NBATCH), 256, 0, stream>>>(inputs, targets, task_id, partials);
  marg_exc_finalize<<<1, 32, 0, stream>>>(partials, task_id, out);
}